// GNNWithMoE_79061757984894
// MI455X (gfx1250) — compile-verified
//
#include <hip/hip_runtime.h>
#include <math.h>

typedef float v2f __attribute__((ext_vector_type(2)));
typedef float v8f __attribute__((ext_vector_type(8)));

#define NDIM 64
#define GDIM 192

__device__ __forceinline__ v8f wmma4(v2f a, v2f b, v8f c) {
  // D = A(16x4 f32) * B(4x16 f32) + C(16x16 f32)
  return __builtin_amdgcn_wmma_f32_16x16x4_f32(false, a, false, b, (short)0, c, false, false);
}

__device__ __forceinline__ float sigmoidf_(float x) {
  return 1.0f / (1.0f + __expf(-x));
}

// Hardware no-return f32 atomic add (GLOBAL_ATOMIC_ADD_F32, STOREcnt-tracked).
// Inline asm guarantees the native L2 atomic (no CAS-loop fallback).
__device__ __forceinline__ void atomic_add_f32(float* p, float v) {
  asm volatile("global_atomic_add_f32 %0, %1, off" : : "v"(p), "v"(v) : "memory");
}

// ---------------- m init: m[i,:] = x[i,:] + b_edge  (self-loop message) -----
__global__ void init_m_kernel(const float* __restrict__ X,
                              const float* __restrict__ b_edge,
                              float* __restrict__ M, long total) {
  long id = (long)blockIdx.x * blockDim.x + threadIdx.x;
  if (id >= total) return;
  M[id] = X[id] + b_edge[id & (NDIM - 1)];
}

// ---------------- fused edge-embed + gather + scatter-add -------------------
// 32 threads per edge; lane p handles dims {2p, 2p+1}.
#define SCATTER_LA 1024
__global__ void scatter_kernel(const float* __restrict__ X,
                               const int* __restrict__ ei,
                               const float* __restrict__ ea,
                               const float* __restrict__ w_edge,  // [64][4]
                               const float* __restrict__ b_edge,  // [64]
                               float* __restrict__ M, int E_) {
  long gid = (long)blockIdx.x * blockDim.x + threadIdx.x;
  int e = (int)(gid >> 5);
  int p = (int)(gid & 31);
  if (e >= E_) return;
  // lookahead prefetch of a future source row (random 256B line, L2-resident)
  int ef = e + SCATTER_LA;
  if (p == 0 && ef < E_) {
    int sf = ei[ef];
    __builtin_prefetch(X + (size_t)sf * NDIM, 0, 0);   // global_prefetch_b8
  }
  int src = ei[e];
  int dst = ei[E_ + e];
  float a0 = ea[e * 4 + 0], a1 = ea[e * 4 + 1];
  float a2 = ea[e * 4 + 2], a3 = ea[e * 4 + 3];
  int d0 = 2 * p;
  const float* w0 = w_edge + d0 * 4;
  float em0 = b_edge[d0]     + a0 * w0[0] + a1 * w0[1] + a2 * w0[2] + a3 * w0[3];
  float em1 = b_edge[d0 + 1] + a0 * w0[4] + a1 * w0[5] + a2 * w0[6] + a3 * w0[7];
  float2 xv = *(const float2*)(X + (size_t)src * NDIM + d0);
  float* mp = M + (size_t)dst * NDIM + d0;
  atomic_add_f32(mp + 0, xv.x + em0);
  atomic_add_f32(mp + 1, xv.y + em1);
}

// ---------------- GRU cell via f32 WMMA, gates fused in registers -----------
// One wave per 16-node tile. For each 16-col block cb we build 6 accumulators
// (gi/gh for r,z,n) with 16 WMMA k-steps each, then combine elementwise.
__global__ __launch_bounds__(128) void gru_kernel(
    const float* __restrict__ Mn,    // [N][64] messages (A for gi)
    const float* __restrict__ X,     // [N][64] hidden state (A for gh)
    const float* __restrict__ w_ih,  // [192][64]
    const float* __restrict__ w_hh,  // [192][64]
    const float* __restrict__ b_ih,  // [192]
    const float* __restrict__ b_hh,  // [192]
    float* __restrict__ out, int ntiles) {
  int lane = threadIdx.x & 31;
  int wave = threadIdx.x >> 5;
  int tile = blockIdx.x * (blockDim.x >> 5) + wave;
  if (tile >= ntiles) return;  // wave-uniform: EXEC all ones below
  int r0 = tile * 16;
  int ml = lane & 15;
  int hi = lane >> 4;
  const float* Am = Mn + (size_t)(r0 + ml) * NDIM;
  const float* Ax = X + (size_t)(r0 + ml) * NDIM;

  for (int cb = 0; cb < 4; ++cb) {
    int c_r = cb * 16 + ml;         // col into [0,192): r block
    int c_z = 64 + cb * 16 + ml;    // z block
    int c_n = 128 + cb * 16 + ml;   // n block
    v8f air = {}, aiz = {}, ain = {}, ahr = {}, ahz = {}, ahn = {};
    const float* bir_p = w_ih + (size_t)c_r * NDIM;
    const float* biz_p = w_ih + (size_t)c_z * NDIM;
    const float* bin_p = w_ih + (size_t)c_n * NDIM;
    const float* bhr_p = w_hh + (size_t)c_r * NDIM;
    const float* bhz_p = w_hh + (size_t)c_z * NDIM;
    const float* bhn_p = w_hh + (size_t)c_n * NDIM;
#pragma unroll
    for (int s = 0; s < 16; ++s) {
      int k0 = 4 * s + 2 * hi;      // per-lane K offset (16x4 f32 A layout)
      v2f am = *(const v2f*)(Am + k0);
      v2f ax = *(const v2f*)(Ax + k0);
      air = wmma4(am, *(const v2f*)(bir_p + k0), air);
      aiz = wmma4(am, *(const v2f*)(biz_p + k0), aiz);
      ain = wmma4(am, *(const v2f*)(bin_p + k0), ain);
      ahr = wmma4(ax, *(const v2f*)(bhr_p + k0), ahr);
      ahz = wmma4(ax, *(const v2f*)(bhz_p + k0), ahz);
      ahn = wmma4(ax, *(const v2f*)(bhn_p + k0), ahn);
    }
    float bir = b_ih[c_r], biz = b_ih[c_z], bin_ = b_ih[c_n];
    float bhr = b_hh[c_r], bhz = b_hh[c_z], bhn = b_hh[c_n];
#pragma unroll
    for (int v = 0; v < 8; ++v) {
      int node = hi * 8 + v;        // C layout: VGPR v, lanes16-31 -> M+8
      float xv = X[(size_t)(r0 + node) * NDIM + cb * 16 + ml];
      float r = sigmoidf_(air[v] + bir + ahr[v] + bhr);
      float z = sigmoidf_(aiz[v] + biz + ahz[v] + bhz);
      float n = tanhf(ain[v] + bin_ + r * (ahn[v] + bhn));
      float h = (1.0f - z) * n + z * xv;
      out[(size_t)(r0 + node) * NDIM + cb * 16 + ml] = fmaxf(h, 0.0f);  // relu
    }
  }
}

// ---------------- soft MoE: gate softmax + 4 expert GEMMs via WMMA ----------
__global__ __launch_bounds__(128) void moe_kernel(
    const float* __restrict__ H,       // [N][64]
    const float* __restrict__ gate_w,  // [4][64]
    const float* __restrict__ gate_b,  // [4]
    const float* __restrict__ exp_w,   // [4][64][64]
    const float* __restrict__ exp_b,   // [4][64]
    float* __restrict__ out, int ntiles) {
  __shared__ float gw_s[4][16][4];     // per-wave gate weights
  int lane = threadIdx.x & 31;
  int wave = threadIdx.x >> 5;
  int tile = blockIdx.x * (blockDim.x >> 5) + wave;
  if (tile >= ntiles) return;
  int r0 = tile * 16;
  int ml = lane & 15;
  int hi = lane >> 4;

  // gate logits: lane computes experts {2*hi, 2*hi+1} for node ml
  const float* hrow = H + (size_t)(r0 + ml) * NDIM;
  float l0 = gate_b[2 * hi], l1 = gate_b[2 * hi + 1];
  for (int d = 0; d < NDIM; ++d) {
    float hv = hrow[d];
    l0 += hv * gate_w[(2 * hi) * NDIM + d];
    l1 += hv * gate_w[(2 * hi + 1) * NDIM + d];
  }
  // softmax over 4 logits: partner lanes exchange via xor-16 shuffle (wave32)
  float m01 = fmaxf(l0, l1);
  float mx = fmaxf(m01, __shfl_xor(m01, 16, 32));
  float e0 = __expf(l0 - mx), e1 = __expf(l1 - mx);
  float s01 = e0 + e1;
  float inv = 1.0f / (s01 + __shfl_xor(s01, 16, 32));
  gw_s[wave][ml][2 * hi] = e0 * inv;
  gw_s[wave][ml][2 * hi + 1] = e1 * inv;
  __builtin_amdgcn_wave_barrier();   // wave-private LDS, DS ops in-order

  for (int cb = 0; cb < 4; ++cb) {
    int col = cb * 16 + ml;
    float oacc[8] = {0, 0, 0, 0, 0, 0, 0, 0};
    for (int e = 0; e < 4; ++e) {
      v8f acc = {};
      const float* Brow = exp_w + ((size_t)e * NDIM + col) * NDIM;
#pragma unroll
      for (int s = 0; s < 16; ++s) {
        int k0 = 4 * s + 2 * hi;
        v2f a = *(const v2f*)(H + (size_t)(r0 + ml) * NDIM + k0);
        v2f b = *(const v2f*)(Brow + k0);
        acc = wmma4(a, b, acc);
      }
      float bias = exp_b[e * NDIM + col];
#pragma unroll
      for (int v = 0; v < 8; ++v) {
        float val = fmaxf(acc[v] + bias, 0.0f);   // relu(expert out)
        oacc[v] += gw_s[wave][hi * 8 + v][e] * val;
      }
    }
#pragma unroll
    for (int v = 0; v < 8; ++v)
      out[(size_t)(r0 + hi * 8 + v) * NDIM + col] = oacc[v];
  }
}

// ---------------- pooling -----------------------------------------------
__global__ void zero_kernel(float* p, int n) {
  int id = blockIdx.x * blockDim.x + threadIdx.x;
  if (id < n) p[id] = 0.0f;
}

__global__ void pool_kernel(const float* __restrict__ H,
                            const int* __restrict__ batch,
                            float* __restrict__ sums, float* __restrict__ cnt,
                            int n) {
  long gid = (long)blockIdx.x * blockDim.x + threadIdx.x;
  int node = (int)(gid >> 6);
  int d = (int)(gid & 63);
  if (node >= n) return;
  int b = batch[node];
  atomic_add_f32(&sums[b * NDIM + d], H[(size_t)node * NDIM + d]);
  if (d == 0) atomic_add_f32(&cnt[b], 1.0f);
}

// ---------------- head: graph_emb + logits ----------------------------------
#define BGR 64
#define PDIM 100
#define NC 10
__global__ void head_kernel(const float* __restrict__ sums,
                            const float* __restrict__ cnt,
                            const float* __restrict__ flag_in,
                            const float* __restrict__ proj_w,  // [100]
                            const float* __restrict__ proj_b,  // [100]
                            const float* __restrict__ fc_w,    // [10][164]
                            const float* __restrict__ fc_b,    // [10]
                            float* __restrict__ out) {
  int id = blockIdx.x * blockDim.x + threadIdx.x;
  if (id < BGR * NDIM) {
    int b = id >> 6;
    float g = sums[id] / fmaxf(cnt[b], 1.0f);
    out[id] = g;                   // graph_emb (output 0)
    out[BGR * NDIM + id] = g;      // graph_emb (output 1)
  } else if (id < BGR * NDIM + BGR * NC) {
    int t = id - BGR * NDIM;
    int b = t / NC, c = t % NC;
    float fl = flag_in[b];
    const float* fr = fc_w + c * (PDIM + NDIM);
    float acc = fc_b[c];
    for (int p = 0; p < PDIM; ++p)
      acc += (fl * proj_w[p] + proj_b[p]) * fr[p];
    float ic = 1.0f / fmaxf(cnt[b], 1.0f);
    for (int d = 0; d < NDIM; ++d)
      acc += sums[b * NDIM + d] * ic * fr[PDIM + d];
    out[2 * BGR * NDIM + t] = acc;  // logits (output 2)
  }
}

extern "C" void kernel_launch(void* const* d_in, const int* in_sizes, int n_in,
                              void* d_out, int out_size, void* d_ws, size_t ws_size,
                              hipStream_t stream) {
  const float* x       = (const float*)d_in[0];
  const int*   ei      = (const int*)d_in[1];
  const float* ea      = (const float*)d_in[2];
  const int*   batch   = (const int*)d_in[3];
  const float* flag_in = (const float*)d_in[4];
  const float* w_edge1 = (const float*)d_in[5];
  const float* b_edge1 = (const float*)d_in[6];
  const float* w_ih1   = (const float*)d_in[7];
  const float* w_hh1   = (const float*)d_in[8];
  const float* b_ih1   = (const float*)d_in[9];
  const float* b_hh1   = (const float*)d_in[10];
  const float* w_edge2 = (const float*)d_in[11];
  const float* b_edge2 = (const float*)d_in[12];
  const float* w_ih2   = (const float*)d_in[13];
  const float* w_hh2   = (const float*)d_in[14];
  const float* b_ih2   = (const float*)d_in[15];
  const float* b_hh2   = (const float*)d_in[16];
  const float* gate_w  = (const float*)d_in[17];
  const float* gate_b  = (const float*)d_in[18];
  const float* exp_w   = (const float*)d_in[19];
  const float* exp_b   = (const float*)d_in[20];
  const float* proj_w  = (const float*)d_in[21];
  const float* proj_b  = (const float*)d_in[22];
  const float* fc_w    = (const float*)d_in[23];
  const float* fc_b    = (const float*)d_in[24];
  float* out = (float*)d_out;

  int N = in_sizes[0] / NDIM;
  int E = in_sizes[1] / 2;
  int ntiles = (N + 15) / 16;

  float* ws = (float*)d_ws;
  float* M    = ws;
  float* hA   = M + (size_t)N * NDIM;
  float* hB   = hA + (size_t)N * NDIM;
  float* sums = hB + (size_t)N * NDIM;
  float* cnt  = sums + BGR * NDIM;

  long totalND = (long)N * NDIM;
  int gN = (int)((totalND + 255) / 256);
  int gE = (int)(((long)E * 32 + 255) / 256);
  int gT = (ntiles + 3) / 4;  // 4 waves/block

  // ---- layer 1 ----
  init_m_kernel<<<gN, 256, 0, stream>>>(x, b_edge1, M, totalND);
  scatter_kernel<<<gE, 256, 0, stream>>>(x, ei, ea, w_edge1, b_edge1, M, E);
  gru_kernel<<<gT, 128, 0, stream>>>(M, x, w_ih1, w_hh1, b_ih1, b_hh1, hA, ntiles);
  // ---- layer 2 ----
  init_m_kernel<<<gN, 256, 0, stream>>>(hA, b_edge2, M, totalND);
  scatter_kernel<<<gE, 256, 0, stream>>>(hA, ei, ea, w_edge2, b_edge2, M, E);
  gru_kernel<<<gT, 128, 0, stream>>>(M, hA, w_ih2, w_hh2, b_ih2, b_hh2, hB, ntiles);
  // ---- MoE ----
  moe_kernel<<<gT, 128, 0, stream>>>(hB, gate_w, gate_b, exp_w, exp_b, hA, ntiles);
  // ---- pool + head ----
  zero_kernel<<<(BGR * NDIM + BGR + 255) / 256, 256, 0, stream>>>(sums, BGR * NDIM + BGR);
  pool_kernel<<<gN, 256, 0, stream>>>(hA, batch, sums, cnt, N);
  head_kernel<<<(BGR * NDIM + BGR * NC + 255) / 256, 256, 0, stream>>>(
      sums, cnt, flag_in, proj_w, proj_b, fc_w, fc_b, out);
}